// GCN_skip_1_layer_default_7567732376250
// MI455X (gfx1250) — compile-verified
//
#include <hip/hip_runtime.h>
#include <hip/hip_bf16.h>

#define NNODES 50000
#define NEDGES 600000
#define DH 128
#define DOUT 40
#define BN_EPS 1e-5f
#define SKIP_COEF 1.0f

typedef __attribute__((ext_vector_type(2))) float v2f;
typedef __attribute__((ext_vector_type(8))) float v8f;

// ---------------- degree / norm ----------------
__global__ void k_deg_init(float* deg, int n) {
    int i = blockIdx.x * blockDim.x + threadIdx.x;
    if (i < n) deg[i] = 1.0f;               // self-loop contribution
}

__global__ void k_deg_edges(const int* __restrict__ dst, float* deg, int e) {
    int i = blockIdx.x * blockDim.x + threadIdx.x;
    if (i < e) atomicAdd(&deg[dst[i]], 1.0f);
}

__global__ void k_deg_rsqrt(float* deg, int n) {
    int i = blockIdx.x * blockDim.x + threadIdx.x;
    if (i < n) deg[i] = rsqrtf(deg[i]);     // deg >= 1 always (self-loops)
}

// ---------------- WMMA f32 GEMM: T[n x DOUTC] = H[n x 128] @ W[128 x DOUTC] ----------
// W is staged once per block into LDS, pre-swizzled into per-lane B-fragment layout:
//   FR[(k4*NT + t)*32 + lane] = { W[k4*4 + khalf][t*16+col16], W[k4*4+khalf+1][...] }
// so the hot loop is: 1 global_load_b64 (A frag) -> NT x { ds_load_b64 ; v_wmma }.
// One wave per 16-row block computes all NT 16-col tiles (A-fragment reuse x NT).
template <int DOUTC>
__global__ __launch_bounds__(128) void k_gemm_wmma(const float* __restrict__ H,
                                                   const float* __restrict__ W,
                                                   float* __restrict__ T, int n) {
    constexpr int NT = (DOUTC + 15) / 16;          // 16-col tiles (8 or 3)
    __shared__ v2f FR[32 * NT * 32];               // 64KB (DH=128) / 24KB (DOUT=40)

    // --- cooperative stage + swizzle of W (zero-pad cols >= DOUTC: branchless tail) ---
    constexpr int TOTAL = 32 * NT * 32;
    for (int idx = threadIdx.x; idx < TOTAL; idx += 128) {
        int lane = idx & 31;
        int t    = (idx >> 5) % NT;
        int k4   = (idx >> 5) / NT;
        int col  = t * 16 + (lane & 15);
        int k    = k4 * 4 + ((lane >> 4) << 1);
        v2f w;
        if ((DOUTC % 16 == 0) || (col < DOUTC)) {
            w.x = W[k * DOUTC + col];
            w.y = W[(k + 1) * DOUTC + col];
        } else {
            w.x = 0.0f; w.y = 0.0f;
        }
        FR[idx] = w;
    }
    __syncthreads();

    const int lane     = threadIdx.x & 31;
    const int rowBlock = blockIdx.x * 4 + (threadIdx.x >> 5);
    if (rowBlock >= (n >> 4)) return;              // after the barrier: safe

    const int khalf = (lane >> 4) << 1;            // A/B K offset per ISA lane layout
    const int mrow  = (rowBlock << 4) + (lane & 15);
    const float* hp = H + (size_t)mrow * DH + khalf;

    v8f acc[NT] = {};
    for (int k4 = 0; k4 < 32; ++k4) {
        v2f a = *(const v2f*)(hp + k4 * 4);        // global_load_b64, reused NT times
        const v2f* fr = &FR[(k4 * NT) * 32 + lane];
        #pragma unroll
        for (int t = 0; t < NT; ++t)
            acc[t] = __builtin_amdgcn_wmma_f32_16x16x4_f32(
                false, a, false, fr[t * 32], (short)0, acc[t], false, false);
    }

    // D 16x16 f32: VGPR v holds row rowBlock*16 + v + (lane>>4)*8, column col
    const int mbase = (rowBlock << 4) + ((lane >> 4) << 3);
    const int col16 = lane & 15;
    #pragma unroll
    for (int t = 0; t < NT; ++t) {
        int col = t * 16 + col16;
        if ((DOUTC % 16 == 0) || (col < DOUTC)) {
            #pragma unroll
            for (int v = 0; v < 8; ++v)
                T[(size_t)(mbase + v) * DOUTC + col] = acc[t][v];
        }
    }
}

// ---------------- accumulator init: bias + self-loop term ----------------
__global__ void k_acc_init(const float* __restrict__ T, const float* __restrict__ dinv,
                           const float* __restrict__ bias, float* __restrict__ acc,
                           int n, int d) {
    int i = blockIdx.x * blockDim.x + threadIdx.x;
    if (i >= n * d) return;
    int node = i / d, f = i - node * d;
    float di = dinv[node];
    acc[i] = bias[f] + T[i] * di * di;
}

// ---------------- edge scatter-add: acc[dst] += T[src] * dinv[src]*dinv[dst] ----------------
#define EDGES_PER_BLOCK 8
__global__ void k_scatter_edges(const float* __restrict__ T, const int* __restrict__ src,
                                const int* __restrict__ dst, const float* __restrict__ dinv,
                                float* __restrict__ acc, int e, int d) {
    int f = threadIdx.x;
    if (f >= d) return;
    int ebase = blockIdx.x * EDGES_PER_BLOCK;
    #pragma unroll
    for (int j = 0; j < EDGES_PER_BLOCK; ++j) {
        int ei = ebase + j;
        if (ei >= e) return;
        int s = src[ei], t = dst[ei];
        float nrm = dinv[s] * dinv[t];
        atomicAdd(&acc[(size_t)t * d + f], T[(size_t)s * d + f] * nrm);
    }
}

// ---------------- BatchNorm stats ----------------
__global__ void k_bn_zero(float* sums) {
    int i = blockIdx.x * blockDim.x + threadIdx.x;
    if (i < 2 * DH) sums[i] = 0.0f;
}

__global__ void k_bn_stats(const float* __restrict__ acc, float* __restrict__ sums, int n) {
    int f = threadIdx.x;                 // 128 threads: one feature each (coalesced rows)
    float s = 0.0f, sq = 0.0f;
    for (int r = blockIdx.x; r < n; r += gridDim.x) {
        float v = acc[(size_t)r * DH + f];
        s += v;
        sq += v * v;
    }
    atomicAdd(&sums[f], s);
    atomicAdd(&sums[DH + f], sq);
}

// ---------------- fused BN affine + optional skip + relu (in place) ----------------
__global__ void k_bn_apply(float* __restrict__ acc, const float* __restrict__ sums,
                           const float* __restrict__ g, const float* __restrict__ be,
                           const float* __restrict__ skip, float skipCoef, int n) {
    int i = blockIdx.x * blockDim.x + threadIdx.x;
    if (i >= n * DH) return;
    int f = i & (DH - 1);
    const float invN = 1.0f / (float)n;
    float mu  = sums[f] * invN;
    float var = sums[DH + f] * invN - mu * mu;      // biased variance (matches jnp var)
    float rstd = rsqrtf(var + BN_EPS);
    float v = g[f] * (acc[i] - mu) * rstd + be[f];
    if (skip) v += skipCoef * skip[i];
    acc[i] = fmaxf(v, 0.0f);
}

// ---------------- driver ----------------
extern "C" void kernel_launch(void* const* d_in, const int* in_sizes, int n_in,
                              void* d_out, int out_size, void* d_ws, size_t ws_size,
                              hipStream_t stream) {
    const float* x   = (const float*)d_in[0];
    const int*   ei  = (const int*)d_in[1];       // [2, E] int32 (JAX x64-off)
    const int*   src = ei;
    const int*   dst = ei + NEDGES;
    const float* W0  = (const float*)d_in[2];
    const float* b0  = (const float*)d_in[3];
    const float* g0  = (const float*)d_in[4];
    const float* be0 = (const float*)d_in[5];
    const float* W1  = (const float*)d_in[6];
    const float* b1  = (const float*)d_in[7];
    const float* g1  = (const float*)d_in[8];
    const float* be1 = (const float*)d_in[9];
    const float* W2  = (const float*)d_in[10];
    const float* b2  = (const float*)d_in[11];
    float* out = (float*)d_out;

    // workspace carve-up (~77MB: fits in MI455X's 192MB L2 -> scatter atomics stay on-chip)
    float* ws   = (float*)d_ws;
    float* dinv = ws;                      // N
    float* sums = ws + NNODES;             // 2*DH
    float* T    = sums + 2 * DH;           // N*DH  (GEMM result)
    float* A    = T + (size_t)NNODES * DH; // N*DH  (conv0 acc -> h1, the skip)
    float* B    = A + (size_t)NNODES * DH; // N*DH  (conv1 acc -> h2)

    const int TPB = 256;
    const int nBlkN   = (NNODES + TPB - 1) / TPB;
    const int nBlkE   = (NEDGES + TPB - 1) / TPB;
    const int nBlkND  = (NNODES * DH + TPB - 1) / TPB;
    const int nBlkNDo = (NNODES * DOUT + TPB - 1) / TPB;
    const int scatBlk = (NEDGES + EDGES_PER_BLOCK - 1) / EDGES_PER_BLOCK;

    // --- GCN normalization (computed once, reused by all 3 convs) ---
    k_deg_init <<<nBlkN, TPB, 0, stream>>>(dinv, NNODES);
    k_deg_edges<<<nBlkE, TPB, 0, stream>>>(dst, dinv, NEDGES);
    k_deg_rsqrt<<<nBlkN, TPB, 0, stream>>>(dinv, NNODES);

    // GEMM geometry: 4 waves/block, one 16-row block per wave covering all col tiles
    const int gemmBlk = ((NNODES / 16) + 3) / 4;   // 782 blocks

    // ---------- layer 0: h1 = relu(bn(conv(x, W0, b0))) ----------
    k_gemm_wmma<DH><<<gemmBlk, 128, 0, stream>>>(x, W0, T, NNODES);
    k_acc_init     <<<nBlkND, TPB, 0, stream>>>(T, dinv, b0, A, NNODES, DH);
    k_scatter_edges<<<scatBlk, DH, 0, stream>>>(T, src, dst, dinv, A, NEDGES, DH);
    k_bn_zero      <<<1, TPB, 0, stream>>>(sums);
    k_bn_stats     <<<512, DH, 0, stream>>>(A, sums, NNODES);
    k_bn_apply     <<<nBlkND, TPB, 0, stream>>>(A, sums, g0, be0, nullptr, 0.0f, NNODES);

    // ---------- layer 1: h2 = relu(bn(conv(h1, W1, b1)) + skip*h1) ----------
    k_gemm_wmma<DH><<<gemmBlk, 128, 0, stream>>>(A, W1, T, NNODES);
    k_acc_init     <<<nBlkND, TPB, 0, stream>>>(T, dinv, b1, B, NNODES, DH);
    k_scatter_edges<<<scatBlk, DH, 0, stream>>>(T, src, dst, dinv, B, NEDGES, DH);
    k_bn_zero      <<<1, TPB, 0, stream>>>(sums);
    k_bn_stats     <<<512, DH, 0, stream>>>(B, sums, NNODES);
    k_bn_apply     <<<nBlkND, TPB, 0, stream>>>(B, sums, g1, be1, A, SKIP_COEF, NNODES);

    // ---------- layer 2: out = conv(h2, W2, b2) ----------
    k_gemm_wmma<DOUT><<<gemmBlk, 128, 0, stream>>>(B, W2, T, NNODES);
    k_acc_init     <<<nBlkNDo, TPB, 0, stream>>>(T, dinv, b2, out, NNODES, DOUT);
    k_scatter_edges<<<scatBlk, 64, 0, stream>>>(T, src, dst, dinv, out, NEDGES, DOUT);
}